// CompressedModel_89687507076174
// MI455X (gfx1250) — compile-verified
//
#include <hip/hip_runtime.h>

#define VEC 8
#define EPS 1e-8f

typedef __attribute__((ext_vector_type(2))) float v2f;
typedef __attribute__((ext_vector_type(8))) float v8f;
typedef __attribute__((ext_vector_type(4))) int   v4i;

// --- gfx1250 async global->LDS path (builtin confirmed present; exact
// param types from the round-2 diagnostic: (v4i AS1*, v4i AS3*, Ii, Ii)) ----
#if defined(__has_builtin)
#  if __has_builtin(__builtin_amdgcn_global_load_async_to_lds_b128) && \
      __has_builtin(__builtin_amdgcn_s_wait_asynccnt)
#    define HAVE_ASYNC_LDS 1
#  endif
#endif
#ifndef HAVE_ASYNC_LDS
#  define HAVE_ASYNC_LDS 0
#endif

#if HAVE_ASYNC_LDS
typedef __attribute__((address_space(1))) v4i* g1_v4i_p;
typedef __attribute__((address_space(3))) v4i* l3_v4i_p;

__device__ __forceinline__ void async_ld_b128(const float* gsrc, float* ldst) {
    __builtin_amdgcn_global_load_async_to_lds_b128(
        (g1_v4i_p)(void*)gsrc,   // strip const, generic->AS1 addrspacecast
        (l3_v4i_p)(void*)ldst,   // generic->AS3 addrspacecast
        0, 0);
}
#endif

// ---------------------------------------------------------------------------
// Kernel 1: VPTQ quantization.
// weight [4096,4096] viewed as [Nv=2097152, 8]; centroids [256,8] (unit-norm).
// One vector per lane; centroids + |c|^2 cached in LDS (broadcast reads).
// ---------------------------------------------------------------------------
__global__ __launch_bounds__(256) void vptq_quantize_kernel(
    const float* __restrict__ weight,
    const float* __restrict__ centroids,
    float* __restrict__ qweight)
{
    __shared__ float sc[256 * VEC];   // centroids
    __shared__ float sc2[256];        // |c_k|^2

    const int tid = threadIdx.x;
    for (int i = tid; i < 256 * VEC; i += 256) sc[i] = centroids[i];
    __syncthreads();
    {
        const float* c = &sc[tid * VEC];
        float s = 0.f;
#pragma unroll
        for (int j = 0; j < VEC; ++j) s = fmaf(c[j], c[j], s);
        sc2[tid] = s;
    }
    __syncthreads();

    const long long vidx = (long long)blockIdx.x * 256 + tid;
    const float4* vp = (const float4*)(weight + vidx * VEC);
    const float4 va = vp[0];
    const float4 vb = vp[1];
    float v[VEC] = {va.x, va.y, va.z, va.w, vb.x, vb.y, vb.z, vb.w};

    float ss = 0.f;
#pragma unroll
    for (int j = 0; j < VEC; ++j) ss = fmaf(v[j], v[j], ss);
    const float inv = 1.0f / (sqrtf(ss) + EPS);
    float nrm[VEC];
#pragma unroll
    for (int j = 0; j < VEC; ++j) nrm[j] = v[j] * inv;

    // argmin_k (|c_k|^2 - 2 n.c_k)  ==  argmin_k ||n - c_k||^2
    int best = 0;
    float bestd = 3.4e38f;
    for (int k = 0; k < 256; ++k) {
        const float* c = &sc[k * VEC];
        float dot = 0.f;
#pragma unroll
        for (int j = 0; j < VEC; ++j) dot = fmaf(nrm[j], c[j], dot);
        const float d = fmaf(-2.f, dot, sc2[k]);
        if (d < bestd) { bestd = d; best = k; }
    }

    const float* a = &sc[best * VEC];
    float num = 0.f;
#pragma unroll
    for (int j = 0; j < VEC; ++j) num = fmaf(v[j], a[j], num);
    const float scale = num / (sc2[best] + EPS);

    float4 o0 = {a[0] * scale, a[1] * scale, a[2] * scale, a[3] * scale};
    float4 o1 = {a[4] * scale, a[5] * scale, a[6] * scale, a[7] * scale};
    float4* qp = (float4*)(qweight + vidx * VEC);
    qp[0] = o0;
    qp[1] = o1;
}

// ---------------------------------------------------------------------------
// Kernel 2: out[M,N] = x[M,K] * qw[N,K]^T + bias, f32 WMMA 16x16x4.
// Workgroup tile 128x128, BK=32, 8 waves (4 along M x 2 along N),
// each wave: 2x4 accumulators of 16x16.
// Async path: double-buffered LDS, GLOBAL_LOAD_ASYNC_TO_LDS_B128 DMA of
// tile k+1 overlapped with WMMA compute of tile k (ASYNCcnt pipelining).
// ---------------------------------------------------------------------------
#define BM 128
#define BN 128
#define BK 32
#define LDSS 36   // padded LDS stride in floats (bank-conflict-free, 16B aligned)

__device__ __forceinline__ void mma_tile(const float* tA, const float* tB,
                                         v8f acc[2][4],
                                         int wave_m, int wave_n, int lane)
{
    const int mrow  = lane & 15;         // fragment row within 16
    const int khalf = (lane >> 4) * 2;   // lanes 0-15: K 0,1; lanes 16-31: K 2,3
#pragma unroll
    for (int ks = 0; ks < 8; ++ks) {
        const int kcol = ks * 4 + khalf;
        v2f afrag[2], bfrag[4];
#pragma unroll
        for (int mi = 0; mi < 2; ++mi)
            afrag[mi] = *(const v2f*)&tA[(wave_m * 32 + mi * 16 + mrow) * LDSS + kcol];
#pragma unroll
        for (int ni = 0; ni < 4; ++ni)
            bfrag[ni] = *(const v2f*)&tB[(wave_n * 64 + ni * 16 + mrow) * LDSS + kcol];
#pragma unroll
        for (int mi = 0; mi < 2; ++mi)
#pragma unroll
            for (int ni = 0; ni < 4; ++ni)
                acc[mi][ni] = __builtin_amdgcn_wmma_f32_16x16x4_f32(
                    /*neg_a=*/false, afrag[mi],
                    /*neg_b=*/false, bfrag[ni],
                    /*c_mod=*/(short)0, acc[mi][ni],
                    /*reuse_a=*/false, /*reuse_b=*/false);
    }
}

__global__ __launch_bounds__(256) void wmma_gemm_kernel(
    const float* __restrict__ x,     // [M, K]
    const float* __restrict__ qw,    // [N, K]
    const float* __restrict__ bias,  // [N]
    float* __restrict__ out,         // [M, N]
    int M, int N, int K)
{
    const int tid    = threadIdx.x;
    const int lane   = tid & 31;
    const int wid    = tid >> 5;
    const int wave_m = wid & 3;          // 0..3 -> 32 rows of M each
    const int wave_n = wid >> 2;         // 0..1 -> 64 cols of N each
    const int block_m = blockIdx.y * BM;
    const int block_n = blockIdx.x * BN;

    v8f acc[2][4];
    {
        v8f zero = {0.f, 0.f, 0.f, 0.f, 0.f, 0.f, 0.f, 0.f};
#pragma unroll
        for (int mi = 0; mi < 2; ++mi)
#pragma unroll
            for (int ni = 0; ni < 4; ++ni) acc[mi][ni] = zero;
    }

    const int ktiles = K / BK;

#if HAVE_ASYNC_LDS
    // ---------------- async DMA path: LDS double buffer -------------------
    __shared__ float sA[2][BM * LDSS];
    __shared__ float sB[2][BN * LDSS];

    // Each wave DMAs 4 A-chunks + 4 B-chunks of 512B (32 lanes x 16B),
    // scattering directly into the padded LDS layout.
    const int crow = (lane >> 3);        // 0..3 row within chunk
    const int ccol = (lane & 7) * 4;     // 0,4,...,28

    // issue tile 0 into buffer 0
#pragma unroll
    for (int i = 0; i < 4; ++i) {
        const int row = (wid * 4 + i) * 4 + crow;   // 0..127
        async_ld_b128(x  + (long long)(block_m + row) * K + ccol,
                      &sA[0][row * LDSS + ccol]);
        async_ld_b128(qw + (long long)(block_n + row) * K + ccol,
                      &sB[0][row * LDSS + ccol]);
    }

    for (int kt = 0; kt < ktiles; ++kt) {
        const int cur = kt & 1;
        if (kt + 1 < ktiles) {
            const int nxt = cur ^ 1;
            const int kc  = (kt + 1) * BK + ccol;
#pragma unroll
            for (int i = 0; i < 4; ++i) {
                const int row = (wid * 4 + i) * 4 + crow;
                async_ld_b128(x  + (long long)(block_m + row) * K + kc,
                              &sA[nxt][row * LDSS + ccol]);
                async_ld_b128(qw + (long long)(block_n + row) * K + kc,
                              &sB[nxt][row * LDSS + ccol]);
            }
            // 8 outstanding belong to tile kt+1; tile kt's DMA has landed
            __builtin_amdgcn_s_wait_asynccnt(8);
        } else {
            __builtin_amdgcn_s_wait_asynccnt(0);
        }
        __syncthreads();                 // tile kt visible to all waves

        mma_tile(&sA[cur][0], &sB[cur][0], acc, wave_m, wave_n, lane);

        __syncthreads();                 // buf cur free before reuse
    }
#else
    // ---------------- fallback: register-staged single buffer -------------
    __shared__ float sA1[BM * LDSS];
    __shared__ float sB1[BN * LDSS];

    const int lrow = tid >> 3;           // 0..31
    const int lcol = (tid & 7) * 4;      // 0,4,...,28

    float4 aReg[4], bReg[4];
#pragma unroll
    for (int p = 0; p < 4; ++p) {
        aReg[p] = *(const float4*)(x  + (long long)(block_m + p * 32 + lrow) * K + lcol);
        bReg[p] = *(const float4*)(qw + (long long)(block_n + p * 32 + lrow) * K + lcol);
    }

    for (int kt = 0; kt < ktiles; ++kt) {
#pragma unroll
        for (int p = 0; p < 4; ++p) {
            *(float4*)&sA1[(p * 32 + lrow) * LDSS + lcol] = aReg[p];
            *(float4*)&sB1[(p * 32 + lrow) * LDSS + lcol] = bReg[p];
        }
        __syncthreads();

        if (kt + 1 < ktiles) {
            const int kc = (kt + 1) * BK + lcol;
#pragma unroll
            for (int p = 0; p < 4; ++p) {
                aReg[p] = *(const float4*)(x  + (long long)(block_m + p * 32 + lrow) * K + kc);
                bReg[p] = *(const float4*)(qw + (long long)(block_n + p * 32 + lrow) * K + kc);
            }
            if (kt + 2 < ktiles) {
                const int kp = (kt + 2) * BK + lcol;
                __builtin_prefetch(x  + (long long)(block_m + lrow) * K + kp, 0, 0);
                __builtin_prefetch(qw + (long long)(block_n + lrow) * K + kp, 0, 0);
            }
        }

        mma_tile(sA1, sB1, acc, wave_m, wave_n, lane);
        __syncthreads();
    }
#endif

    // Epilogue: C/D layout — VGPR j holds M=j (lanes 0-15) and M=j+8 (lanes 16-31),
    // N = lane % 16. Fuse bias add.
    const int mhi  = (lane >> 4) * 8;
    const int ncol = lane & 15;
#pragma unroll
    for (int ni = 0; ni < 4; ++ni) {
        const int n0 = block_n + wave_n * 64 + ni * 16;
        const float bval = bias[n0 + ncol];
#pragma unroll
        for (int mi = 0; mi < 2; ++mi) {
            const int m0 = block_m + wave_m * 32 + mi * 16;
#pragma unroll
            for (int j = 0; j < 8; ++j) {
                out[(long long)(m0 + j + mhi) * N + n0 + ncol] = acc[mi][ni][j] + bval;
            }
        }
    }
}

// ---------------------------------------------------------------------------
// Launch
// ---------------------------------------------------------------------------
extern "C" void kernel_launch(void* const* d_in, const int* in_sizes, int n_in,
                              void* d_out, int out_size, void* d_ws, size_t ws_size,
                              hipStream_t stream) {
    (void)in_sizes; (void)n_in; (void)out_size; (void)ws_size;

    const float* x         = (const float*)d_in[0];  // [4,4096,4096]
    const float* weight    = (const float*)d_in[1];  // [4096,4096]
    const float* bias      = (const float*)d_in[2];  // [4096]
    const float* centroids = (const float*)d_in[3];  // [256,8]
    float* out = (float*)d_out;                      // [4,4096,4096]
    float* qweight = (float*)d_ws;                   // 64 MB scratch

    const int M = 4 * 4096;
    const int N = 4096;
    const int K = 4096;
    const int NV = (N / VEC) * K;                    // 2,097,152 vectors

    vptq_quantize_kernel<<<NV / 256, 256, 0, stream>>>(weight, centroids, qweight);

    dim3 grid(N / BN, M / BM);
    wmma_gemm_kernel<<<grid, dim3(256), 0, stream>>>(x, qweight, bias, out, M, N, K);
}